// Linear_59236188946489
// MI455X (gfx1250) — compile-verified
//
#include <hip/hip_runtime.h>

// CDNA5 / gfx1250: wave32, WMMA (V_WMMA_F32_16X16X4_F32 for the f32 GEMM).
typedef __attribute__((ext_vector_type(2))) float v2f;
typedef __attribute__((ext_vector_type(8))) float v8f;

#define XS_STRIDE 836   // 832 + pad: bank stride 4; half-wave offset 2d (d odd) ≡ 2 mod 4 → conflict-free
#define WSP01 144       // row-pair stride (float2 units), 128 cols; ≡16 mod 32 → halves hit disjoint 32-bank ranges
#define WSP2  80        // row-pair stride (float2 units), 64 cols; ≡16 mod 32
#define TROWS 32        // x rows staged per block (N = 2^17 divisible; all tile counts integral)
#define THREADS 512     // 16 wave32s

__device__ __forceinline__ v8f wmma_f32(v2f a, v2f b, v8f c) {
  // D = A(16x4) * B(4x16) + C, f32 in/out:
  // (neg_a, A, neg_b, B, c_mod, C, reuse_a, reuse_b)
  return __builtin_amdgcn_wmma_f32_16x16x4_f32(false, a, false, b, (short)0, c,
                                               false, false);
}

// One 16-combined-row x MUL-col tile of segment <MUL, D, OFF>.
// W staged in LDS pre-interleaved (and pre-scaled by alpha):
//   wp[p*WSP + v] = alpha * {W[2p][v], W[2p+1][v]}
// so a B fragment (rows u,u+1 at col v; u = 4kk+2*half even, p = 2kk+half)
// is a single ds_load_b64.  f32 16x4 A layout: lane L (<16) holds
// A[M=L][k],[k+1]; lane L+16 holds A[M=L][k+2],[k+3].
// Stores: uniform block base (SGPR) + 32-bit voffset; the per-c column term
// (c*64*D bytes) is a compile-time constant folded into the 24-bit imm offset.
template<int MUL, int D, int OFF, int WSP>
__device__ __forceinline__ void process_tile(const v2f* wp,
                                             const float* xs,
                                             float* __restrict__ outb,
                                             int rt, int lane)
{
  constexpr int NK  = MUL / 4;    // K-chunks of 4
  constexpr int NCT = MUL / 16;   // 16-wide output column tiles
  const int lm   = lane & 15;
  const int half = lane >> 4;
  const int rbase = rt * 16;

  // A-fragment source row for this lane (both half-waves use M = lm)
  const int rA = rbase + lm;
  const int nA = rA / D;
  const int iA = rA - nA * D;
  const int abase = nA * XS_STRIDE + OFF + iA;
  const int kb = half * 2;        // lanes 16-31 take K+2, K+3

  v8f acc[NCT];
  v8f zero = {0.f, 0.f, 0.f, 0.f, 0.f, 0.f, 0.f, 0.f};
#pragma unroll
  for (int c = 0; c < NCT; ++c) acc[c] = zero;

#pragma unroll 2
  for (int kk = 0; kk < NK; ++kk) {
    const int u = kk * 4 + kb;
    v2f A;
    A.x = xs[abase + u * D];               // ds_load_2addr (offsets 0, D)
    A.y = xs[abase + (u + 1) * D];
    const v2f* wr = wp + (kk * 2 + half) * WSP + lm;  // one base add per kk
#pragma unroll
    for (int c = 0; c < NCT; ++c) {
      v2f B = wr[c * 16];                  // ds_load_b64
      acc[c] = wmma_f32(A, B, acc[c]);
    }
  }

  // D layout: VGPR j -> row M = j + 8*half, col N = lm.
  // voff[j] (byte offsets for the 8 rows) computed once per tile with a
  // 32-bit incremental carry; shared across all column tiles.
  const int r0 = rbase + half * 8;
  const int nS = r0 / D;
  const int iS = r0 - nS * D;
  unsigned voff[8];
  {
    unsigned d = (unsigned)(nS * 832 + OFF + iS + lm * D) * 4u;
    int ii = iS;
#pragma unroll
    for (int j = 0; j < 8; ++j) {
      voff[j] = d;
      if (D == 1) {
        d += 832u * 4u;
      } else if (ii == D - 1) {
        ii = 0;
        d += (unsigned)(833 - D) * 4u;
      } else {
        ++ii;
        d += 4u;
      }
    }
  }
#pragma unroll
  for (int c = 0; c < NCT; ++c) {
#pragma unroll
    for (int j = 0; j < 8; ++j) {
      *(float*)((char*)outb + (size_t)voff[j] + (size_t)(c * 64 * D)) =
          acc[c][j];
    }
  }
}

__global__ __launch_bounds__(THREADS, 1)
void seg_linear_kernel(const float* __restrict__ x,
                       const float* __restrict__ W0,
                       const float* __restrict__ W1,
                       const float* __restrict__ W2,
                       float* __restrict__ out)
{
  // 107,008 + 73,728 + 73,728 + 20,480 = 274,944 B < 320 KB/WGP
  __shared__ __attribute__((aligned(16))) float xs[TROWS * XS_STRIDE];
  __shared__ __attribute__((aligned(16))) v2f  wp0[64 * WSP01];
  __shared__ __attribute__((aligned(16))) v2f  wp1[64 * WSP01];
  __shared__ __attribute__((aligned(16))) v2f  wp2[32 * WSP2];

  const int n0  = blockIdx.x * TROWS;
  const int tid = threadIdx.x;

  // Coalesced stage of 32 rows x 832 f32 (contiguous in HBM) into padded LDS.
  const float4* src = (const float4*)(x + (size_t)n0 * 832);
  for (int idx = tid; idx < TROWS * 208; idx += THREADS) {
    const int row = idx / 208;
    const int c4  = idx - row * 208;
    float4 val = src[(size_t)row * 208 + c4];
    *(float4*)&xs[row * XS_STRIDE + c4 * 4] = val;
  }

  // Stage weights (L2-hot) into LDS: interleave row pairs into float2 and
  // pre-scale by the e3nn path norm alpha = 1/sqrt(fan_in), matching the
  // reference's W*alpha.
  {
    const float a01 = 0.08838834764831845f;  // 1/sqrt(128)
    const float a2  = 0.125f;                // 1/sqrt(64)
    const float4* s0 = (const float4*)W0;
    const float4* s1 = (const float4*)W1;
    for (int idx = tid; idx < 64 * 32; idx += THREADS) {   // p in [0,64), v4 in [0,32)
      const int p  = idx >> 5;
      const int v4 = idx & 31;
      float4 a0 = s0[(p * 2) * 32 + v4];
      float4 b0 = s0[(p * 2 + 1) * 32 + v4];
      float4 a1 = s1[(p * 2) * 32 + v4];
      float4 b1 = s1[(p * 2 + 1) * 32 + v4];
      v2f* d0 = &wp0[p * WSP01 + v4 * 4];
      v2f* d1 = &wp1[p * WSP01 + v4 * 4];
      d0[0] = v2f{a0.x * a01, b0.x * a01}; d0[1] = v2f{a0.y * a01, b0.y * a01};
      d0[2] = v2f{a0.z * a01, b0.z * a01}; d0[3] = v2f{a0.w * a01, b0.w * a01};
      d1[0] = v2f{a1.x * a01, b1.x * a01}; d1[1] = v2f{a1.y * a01, b1.y * a01};
      d1[2] = v2f{a1.z * a01, b1.z * a01}; d1[3] = v2f{a1.w * a01, b1.w * a01};
    }
    const float4* s2 = (const float4*)W2;
    for (int idx = tid; idx < 32 * 16; idx += THREADS) {   // p in [0,32), v4 in [0,16)
      const int p  = idx >> 4;
      const int v4 = idx & 15;
      float4 a = s2[(p * 2) * 16 + v4];
      float4 b = s2[(p * 2 + 1) * 16 + v4];
      v2f* d = &wp2[p * WSP2 + v4 * 4];
      d[0] = v2f{a.x * a2, b.x * a2}; d[1] = v2f{a.y * a2, b.y * a2};
      d[2] = v2f{a.z * a2, b.z * a2}; d[3] = v2f{a.w * a2, b.w * a2};
    }
  }
  __syncthreads();

  const int wave = tid >> 5;   // wave32
  const int lane = tid & 31;
  float* outb = out + (size_t)n0 * 832;   // uniform block base (SGPR pair)

  // Per block (32 rows): 2 seg0 tiles + 6 seg1 tiles (256 WMMA each) and
  // 10 seg2 tiles (64 WMMA each).  Waves 0-7: one big tile; waves 8-15:
  // round-robin smalls.  Per-SIMD load is near-even; no divergence in-wave.
  if (wave < 2) {
    process_tile<128, 1, 0, WSP01>(wp0, xs, outb, wave, lane);
  } else if (wave < 8) {
    process_tile<128, 3, 128, WSP01>(wp1, xs, outb, wave - 2, lane);
  } else {
    for (int s = wave - 8; s < 10; s += 8) {
      process_tile<64, 5, 512, WSP2>(wp2, xs, outb, s, lane);
    }
  }
}

extern "C" void kernel_launch(void* const* d_in, const int* in_sizes, int n_in,
                              void* d_out, int out_size, void* d_ws, size_t ws_size,
                              hipStream_t stream) {
  (void)n_in; (void)out_size; (void)d_ws; (void)ws_size;
  const float* x  = (const float*)d_in[0];
  const float* W0 = (const float*)d_in[1];
  const float* W1 = (const float*)d_in[2];
  const float* W2 = (const float*)d_in[3];
  float* out = (float*)d_out;

  const int N = in_sizes[0] / 832;        // 131072
  const int nblocks = N / TROWS;          // 4096
  seg_linear_kernel<<<nblocks, THREADS, 0, stream>>>(x, W0, W1, W2, out);
}